// InChIEncoder_89008902242912
// MI455X (gfx1250) — compile-verified
//
#include <hip/hip_runtime.h>

typedef __attribute__((ext_vector_type(2))) float v2f;
typedef __attribute__((ext_vector_type(8))) float v8f;

#define EMB   256
#define LM1   256
#define NCH   257
#define BATCH 8

// ---------------------------------------------------------------------------
// Prep: exclusive prefix sums of dense_w.  P[j] = sum_{w<j} dense_w[w],
// j = 0..256 (257 floats in workspace).
// ---------------------------------------------------------------------------
__global__ __launch_bounds__(256) void prep_prefix(const float* __restrict__ dense_w,
                                                   float* __restrict__ P) {
  __shared__ float s[256];
  const int t = threadIdx.x;
  s[t] = dense_w[t];
  __syncthreads();
  // Hillis-Steele inclusive scan
  for (int off = 1; off < 256; off <<= 1) {
    float v = (t >= off) ? s[t - off] : 0.0f;
    __syncthreads();
    s[t] += v;
    __syncthreads();
  }
  if (t == 0) P[0] = 0.0f;
  P[t + 1] = s[t];   // inclusive sum through t == exclusive prefix at t+1
}

// ---------------------------------------------------------------------------
// Main kernel: one block per (batch b, output row r), 256 threads (8 waves).
// r == 0      -> copy start_var.
// r == c+1    -> folded conv+contract:
//   z[b,c,e] = relu( S1*relu(Kf.v + bb) + w0*relu(A0.v + bb)
//                  + w2*relu(Ac.v + bb) + w3*relu(Acp.v + bb)
//                  + S2*relu(bb) + dense_b )
//   with v = (x[e-1], x[e], x[e+1]) (zero padded), coefficients folded from
//   the 3x3 depthwise kernel and the tril mask, S1/S2 from prefix sums.
// The four dot-3s are computed with V_WMMA_F32_16X16X4_F32:
//   A (16x4, f32): rows = e positions, K = (xm, x0, xp, 0)
//   B (4x16, f32): columns 0..3 = coefficient sets Kf, A0, Ac, Acp
//   D (16x16):     D[m][n] = dot(coef_n, v_m)
// ---------------------------------------------------------------------------
__global__ __launch_bounds__(256) void inchi_fused(
    const int*   __restrict__ tokens,     // (8, 257)
    const float* __restrict__ emb,        // (128, 256)
    const float* __restrict__ dwk,        // (256, 1, 3, 3)
    const float* __restrict__ dwb,        // (256,)
    const float* __restrict__ dense_w,    // (256,)
    const float* __restrict__ dense_b,    // (1,)
    const float* __restrict__ start_var,  // (256,)
    const float* __restrict__ P,          // (257,) prefix sums
    float*       __restrict__ out)        // (8, 257, 256)
{
  __shared__ float xrow[EMB + 2];   // zero-padded emb row: xrow[1+e] = x[e]
  __shared__ float coef[4][3];      // [set n][e-tap i]
  __shared__ float dmat[8][32][4];  // [wave][e within wave][set n]

  const int blk = blockIdx.x;
  const int b   = blk / NCH;
  const int r   = blk % NCH;
  const int tid = threadIdx.x;

  if (r == 0) {  // start row (block-uniform branch)
    out[(size_t)b * NCH * EMB + tid] = start_var[tid];
    return;
  }
  const int c   = r - 1;
  const int tok = tokens[b * NCH + c];

  // Stage embedding row into LDS with zero halo (conv 'SAME' padding in H).
  xrow[tid + 1] = emb[tok * EMB + tid];
  if (tid == 0) { xrow[0] = 0.0f; xrow[EMB + 1] = 0.0f; }

  // Fold the 3x3 kernel into 4 regime coefficient sets over the 3 e-taps.
  if (tid < 12) {
    const int i = tid >> 2;        // e-tap: 0..2 (rows of 3x3 kernel)
    const int n = tid & 3;         // regime set: 0..3
    const float k0 = dwk[c * 9 + i * 3 + 0];
    const float k1 = dwk[c * 9 + i * 3 + 1];
    const float k2 = dwk[c * 9 + i * 3 + 2];
    float v;
    if      (n == 0) v = k0 + k1 + k2;                   // interior 1..c-1
    else if (n == 1) v = k1 + ((c >= 1) ? k2 : 0.0f);    // w = 0
    else if (n == 2) v = k0 + k1;                        // w = c
    else             v = k0;                             // w = c+1
    coef[n][i] = v;
  }
  __syncthreads();

  const int lane = tid & 31;
  const int wave = tid >> 5;
  const int half = lane >> 4;    // 0: lanes 0-15, 1: lanes 16-31
  const int mrow = lane & 15;    // A-matrix row / B,C,D column index
  const int nn   = (mrow < 4) ? mrow : 0;

  // B matrix (4x16): lanes 0-15 hold K={0,1}, lanes 16-31 hold K={2,3}
  // (mirrors the documented 16x4 A layout half-lane K split).
  float b0 = coef[nn][half ? 2 : 0];
  float b1 = coef[nn][1];
  if (mrow >= 4) { b0 = 0.0f; b1 = 0.0f; }
  if (half)      { b1 = 0.0f; }              // K=3 row is zero padding
  v2f bmat; bmat.x = b0; bmat.y = b1;

#pragma unroll
  for (int h = 0; h < 2; ++h) {              // two 16-row WMMAs cover 32 e's
    const int e = wave * 32 + h * 16 + mrow;
    // A matrix (16x4): lanes 0-15: VGPR0=K0(xm), VGPR1=K1(x0);
    //                  lanes 16-31: VGPR0=K2(xp), VGPR1=K3(0)
    v2f a;
    a.x = xrow[e + (half ? 2 : 0)];          // xm or xp
    a.y = half ? 0.0f : xrow[e + 1];         // x0 or pad
    v8f cz = {};
    v8f d = __builtin_amdgcn_wmma_f32_16x16x4_f32(
        false, a, false, bmat, (short)0, cz, false, false);
    // D layout: VGPR rr -> M = rr + 8*half, N = mrow.  Keep columns 0..3.
    if (mrow < 4) {
#pragma unroll
      for (int rr = 0; rr < 8; ++rr) {
        dmat[wave][h * 16 + rr + (half ? 8 : 0)][mrow] = d[rr];
      }
    }
  }
  __syncthreads();

  // Per-element combine; all weights are uniform per block (per c).
  const float bb = dwb[c];
  const float db = dense_b[0];
  const float S1 = P[(c < 1) ? 1 : c] - P[1];                    // sum w=1..c-1
  const float S2 = (c <= 254) ? (P[256] - P[c + 2]) : 0.0f;      // sum w=c+2..255
  const float w0 = dense_w[0];
  const float w2 = (c >= 1)   ? dense_w[c]     : 0.0f;           // w=c regime
  const float w3 = (c <= 254) ? dense_w[c + 1] : 0.0f;           // w=c+1 regime

  const float acc = S1 * fmaxf(dmat[wave][lane][0] + bb, 0.0f)
                  + w0 * fmaxf(dmat[wave][lane][1] + bb, 0.0f)
                  + w2 * fmaxf(dmat[wave][lane][2] + bb, 0.0f)
                  + w3 * fmaxf(dmat[wave][lane][3] + bb, 0.0f)
                  + S2 * fmaxf(bb, 0.0f) + db;

  out[(size_t)b * NCH * EMB + (size_t)r * EMB + tid] = fmaxf(acc, 0.0f);
}

// ---------------------------------------------------------------------------
extern "C" void kernel_launch(void* const* d_in, const int* in_sizes, int n_in,
                              void* d_out, int out_size, void* d_ws, size_t ws_size,
                              hipStream_t stream) {
  const int*   tokens    = (const int*)  d_in[0];
  const float* emb       = (const float*)d_in[1];
  const float* dw_kernel = (const float*)d_in[2];
  const float* dw_bias   = (const float*)d_in[3];
  const float* dense_w   = (const float*)d_in[4];
  const float* dense_b   = (const float*)d_in[5];
  const float* start_var = (const float*)d_in[6];
  float* out = (float*)d_out;
  float* P   = (float*)d_ws;   // 257 floats

  prep_prefix<<<1, 256, 0, stream>>>(dense_w, P);
  inchi_fused<<<BATCH * NCH, 256, 0, stream>>>(
      tokens, emb, dw_kernel, dw_bias, dense_w, dense_b, start_var, P, out);
}